// SpatialDerivativeOperator_27857157882280
// MI455X (gfx1250) — compile-verified
//
#include <hip/hip_runtime.h>
#include <hip/hip_bf16.h>

// Graph spatial-derivative: edge_deriv = (x[dst]-x[src])/edge_attr,
// node_deriv = scatter_mean(edge_deriv, dst). Pure bandwidth/atomic-bound
// (~260 MB moved, ~77 MFLOP, AI~0.3) -> WMMA is inapplicable; the MI455X
// path that matters is data movement: NT b128 streams (don't evict L2),
// L2-resident global_atomic_add_f32 scatter, global_prefetch_b8 lookahead.

typedef float v4f __attribute__((ext_vector_type(4)));

#define N_NODES 100000
#define N_EDGES 3200000
#define D 8

__global__ __launch_bounds__(256) void
sdo_edge_kernel(const float* __restrict__ x,
                const long long* __restrict__ src_idx,
                const long long* __restrict__ dst_idx,
                const float* __restrict__ edge_attr,
                float* __restrict__ edge_deriv,   // [E, D]
                float* __restrict__ node_sum,     // [N, D] pre-zeroed (d_out region)
                float* __restrict__ node_cnt)     // [N]    pre-zeroed (d_ws)
{
    const int stride = gridDim.x * blockDim.x;
    for (int e = blockIdx.x * blockDim.x + threadIdx.x; e < N_EDGES; e += stride) {
        const size_t row = (size_t)e * D;

        // Prefetch next grid-stride tile of the dominant stream into GL2.
        const int en = e + stride;
        if (en < N_EDGES) {
            __builtin_prefetch(edge_attr + (size_t)en * D, 0, 0);
        }

        // Node ids < 100000 -> 32-bit index math (avoids 64-bit VALU chains).
        const unsigned s = (unsigned)src_idx[e];
        const unsigned d = (unsigned)dst_idx[e];

        // x rows are 32B (2 x b128); x is 3.2MB -> L2-resident gathers (RT).
        const v4f* xs = (const v4f*)(x + (size_t)s * D);
        const v4f* xd = (const v4f*)(x + (size_t)d * D);
        const v4f s0 = xs[0];
        const v4f s1 = xs[1];
        const v4f d0 = xd[0];
        const v4f d1 = xd[1];

        // edge_attr is streamed once: non-temporal to protect L2 residency
        // of x / node_sum / node_cnt.
        const v4f* ea = (const v4f*)(edge_attr + row);
        const v4f a0 = __builtin_nontemporal_load(ea);
        const v4f a1 = __builtin_nontemporal_load(ea + 1);

        const v4f r0 = (d0 - s0) / a0;
        const v4f r1 = (d1 - s1) / a1;

        // Streaming 102MB output: non-temporal b128 stores.
        v4f* out = (v4f*)(edge_deriv + row);
        __builtin_nontemporal_store(r0, out);
        __builtin_nontemporal_store(r1, out + 1);

        // Scatter-sum into L2-resident accumulators: non-returning
        // global_atomic_add_f32 (results discarded -> no-return form).
        float* acc = node_sum + (size_t)d * D;
        unsafeAtomicAdd(acc + 0, r0.x);
        unsafeAtomicAdd(acc + 1, r0.y);
        unsafeAtomicAdd(acc + 2, r0.z);
        unsafeAtomicAdd(acc + 3, r0.w);
        unsafeAtomicAdd(acc + 4, r1.x);
        unsafeAtomicAdd(acc + 5, r1.y);
        unsafeAtomicAdd(acc + 6, r1.z);
        unsafeAtomicAdd(acc + 7, r1.w);
        unsafeAtomicAdd(node_cnt + d, 1.0f);
    }
}

__global__ __launch_bounds__(256) void
sdo_normalize_kernel(float* __restrict__ node,       // [N, D] in-place sums -> means
                     const float* __restrict__ cnt)  // [N]
{
    const int i = blockIdx.x * blockDim.x + threadIdx.x;
    if (i >= N_NODES) return;
    const float inv = 1.0f / fmaxf(cnt[i], 1.0f);
    v4f* p = (v4f*)(node + (size_t)i * D);
    const v4f a = p[0];
    const v4f b = p[1];
    p[0] = a * inv;
    p[1] = b * inv;
}

extern "C" void kernel_launch(void* const* d_in, const int* in_sizes, int n_in,
                              void* d_out, int out_size, void* d_ws, size_t ws_size,
                              hipStream_t stream) {
    const float*     x         = (const float*)d_in[0];
    const long long* edge_idx  = (const long long*)d_in[1];  // [2, E] int64
    const float*     edge_attr = (const float*)d_in[2];

    const long long* src_idx = edge_idx;            // row 0
    const long long* dst_idx = edge_idx + N_EDGES;  // row 1

    // d_out layout (return order): node_deriv [N,D] then edge_deriv [E,D]
    float* node_out = (float*)d_out;
    float* edge_out = (float*)d_out + (size_t)N_NODES * D;

    float* node_cnt = (float*)d_ws;  // [N] floats (400 KB scratch)

    // Zero the accumulator regions every call (graph-capture-safe async memset).
    hipMemsetAsync(node_out, 0, (size_t)N_NODES * D * sizeof(float), stream);
    hipMemsetAsync(node_cnt, 0, (size_t)N_NODES * sizeof(float), stream);

    // Grid-stride edge pass: ~1M threads (8 wave32 waves/block), ~3 edges per
    // thread so the edge_attr prefetch has a live target.
    const int block = 256;
    const int grid_edges = 4096;
    sdo_edge_kernel<<<grid_edges, block, 0, stream>>>(
        x, src_idx, dst_idx, edge_attr, edge_out, node_out, node_cnt);

    const int grid_nodes = (N_NODES + block - 1) / block;
    sdo_normalize_kernel<<<grid_nodes, block, 0, stream>>>(node_out, node_cnt);
}